// WaveletTransform_57294863729399
// MI455X (gfx1250) — compile-verified
//
#include <hip/hip_runtime.h>
#include <cstdint>
#include <cstddef>

// Haar (db1) 2-D DWT, non-overlapping 2x2 block transform for even H,W.
// x: (16,3,1024,1024) f32  ->  d_out = concat(cA,cH,cV,cD), each (16,3,512,512) f32.
//
// Roofline: ~402 MB total traffic -> ~17us floor at 23.3 TB/s; ~151 MFLOP total,
// so this is purely bandwidth-bound (WMMA irrelevant). Data path uses the CDNA5
// async global->LDS streaming engine (ASYNCcnt) with non-temporal hints on both
// the read-once input stream and the write-once subband outputs (working set
// 402 MB > 192 MB L2, nothing is reused).

typedef __attribute__((ext_vector_type(2))) float v2f;
typedef __attribute__((ext_vector_type(4))) float v4f;

#define N_IMG 48            // 16 batches * 3 channels
#define IN_W  1024
#define IN_H  1024
#define OUT_W 512
#define OUT_H 512
#define NQ    (N_IMG * OUT_H * OUT_W)   // elements per subband = 12,582,912

__global__ __launch_bounds__(256)
void haar_dwt2_f32_kernel(const float* __restrict__ x, float* __restrict__ out)
{
    // 8 KB staging buffer: slots [0..255] = row 2r, [256..511] = row 2r+1.
    // Each thread owns exactly its own 16-byte slot in each half -> no barriers,
    // only the per-wave ASYNCcnt wait is needed before readback.
    __shared__ uint4 sh[512];

    const uint32_t t   = threadIdx.x;        // 0..255, covers input cols 4t..4t+3
    const uint32_t bb  = blockIdx.x;         // 0..24575
    const uint32_t r   = bb & 511u;          // output row (input rows 2r, 2r+1)
    const uint32_t img = bb >> 9;            // 0..47 (b*3 + c)

    // ---- Input addresses (fully coalesced: lane stride 16 B -> 512 B/wave) ----
    const uint64_t in_base = ((uint64_t)img * IN_H + 2u * r) * IN_W + (uint64_t)t * 4u;
    const uint64_t ga0 = (uint64_t)(uintptr_t)(x + in_base);             // row 2r
    const uint64_t ga1 = ga0 + (uint64_t)IN_W * sizeof(float);           // row 2r+1

    // LDS byte offsets of this lane's slots (low 32 bits of the generic address
    // of a __shared__ object are its offset within the group segment).
    const uint32_t l0 = (uint32_t)(uintptr_t)(&sh[t]);
    const uint32_t l1 = (uint32_t)(uintptr_t)(&sh[256u + t]);

    // ---- CDNA5 async streaming: global -> LDS, ASYNCcnt-tracked, NT hint ----
    asm volatile("global_load_async_to_lds_b128 %0, %1, off th:TH_LOAD_NT"
                 :: "v"(l0), "v"(ga0) : "memory");
    asm volatile("global_load_async_to_lds_b128 %0, %1, off th:TH_LOAD_NT"
                 :: "v"(l1), "v"(ga1) : "memory");
    asm volatile("s_wait_asynccnt 0" ::: "memory");

    // Readback own slots (ds_load_b128 each).
    const v4f a = *reinterpret_cast<const v4f*>(&sh[t]);          // row 2r  : x00 x01 (2 blocks)
    const v4f b = *reinterpret_cast<const v4f*>(&sh[256u + t]);   // row 2r+1: x10 x11

    // ---- 2x2 Haar butterfly (two blocks per thread) ----
    const float s0 = a.x + a.y, d0 = a.x - a.y;   // block0, row0
    const float s1 = b.x + b.y, d1 = b.x - b.y;   // block0, row1
    const float s2 = a.z + a.w, d2 = a.z - a.w;   // block1, row0
    const float s3 = b.z + b.w, d3 = b.z - b.w;   // block1, row1

    const v2f cA = { 0.5f * (s0 + s1), 0.5f * (s2 + s3) };  // 'aa'
    const v2f cH = { 0.5f * (s0 - s1), 0.5f * (s2 - s3) };  // 'da'
    const v2f cV = { 0.5f * (d0 + d1), 0.5f * (d2 + d3) };  // 'ad'
    const v2f cD = { 0.5f * (d0 - d1), 0.5f * (d2 - d3) };  // 'dd'

    // ---- Non-temporal, fully coalesced b64 stores (write-once data) ----
    const uint64_t q = ((uint64_t)img * OUT_H + r) * OUT_W + (uint64_t)t * 2u;
    __builtin_nontemporal_store(cA, reinterpret_cast<v2f*>(out + q));
    __builtin_nontemporal_store(cH, reinterpret_cast<v2f*>(out + (uint64_t)NQ + q));
    __builtin_nontemporal_store(cV, reinterpret_cast<v2f*>(out + 2ull * NQ + q));
    __builtin_nontemporal_store(cD, reinterpret_cast<v2f*>(out + 3ull * NQ + q));
}

extern "C" void kernel_launch(void* const* d_in, const int* in_sizes, int n_in,
                              void* d_out, int out_size, void* d_ws, size_t ws_size,
                              hipStream_t stream)
{
    const float* x = (const float*)d_in[0];   // (16,3,1024,1024) f32
    float* out     = (float*)d_out;           // 4 * 48*512*512 f32

    (void)in_sizes; (void)n_in; (void)out_size; (void)d_ws; (void)ws_size;

    const dim3 grid(N_IMG * OUT_H);  // 48 * 512 = 24576 blocks
    const dim3 block(256);           // 8 wave32 waves per block
    haar_dwt2_f32_kernel<<<grid, block, 0, stream>>>(x, out);
}